// GPRLayer_21019569947063
// MI455X (gfx1250) — compile-verified
//
#include <hip/hip_runtime.h>
#include <hip/hip_bf16.h>

typedef float v2f __attribute__((ext_vector_type(2)));
typedef float v8f __attribute__((ext_vector_type(8)));

#define D 128
#define BN_EPS 1e-5f

__device__ __forceinline__ void atomic_add_f32(float* p, float v) {
    (void)__hip_atomic_fetch_add(p, v, __ATOMIC_RELAXED, __HIP_MEMORY_SCOPE_AGENT);
}

// ---------------- generic zero ----------------
__global__ void zero_k(float* __restrict__ p, int n) {
    int i = blockIdx.x * blockDim.x + threadIdx.x;
    if (i < n) p[i] = 0.0f;
}

// ---------------- GEMM1: H = X @ W1 + b1  (fp32 WMMA 16x16x4) ----------------
// grid.x = N/16 row tiles, block = 256 (8 waves); wave w does column tile w.
__global__ void gemm1_k(const float* __restrict__ X, const float* __restrict__ W,
                        const float* __restrict__ bias, float* __restrict__ H) {
    const int wave = threadIdx.x >> 5;
    const int lane = threadIdx.x & 31;
    const int row  = lane & 15;     // M index for A loads, N index for B/C
    const int hi   = lane >> 4;     // lane-half select
    const int m0   = blockIdx.x * 16;
    const int n0   = wave * 16;

    v8f acc = {};
    #pragma unroll
    for (int k0 = 0; k0 < D; k0 += 4) {
        // A: 16x4 fp32 tile. lane(l<16): M=l, K={k0,k0+1}; lane(l>=16): M=l-16, K={k0+2,k0+3}
        v2f a = *(const v2f*)&X[(size_t)(m0 + row) * D + k0 + 2 * hi];
        // B: 4x16 fp32 tile. VGPR0: rows K=k0 (lanes 0-15) / K=k0+2 (lanes 16-31); VGPR1: +1
        v2f b;
        b.x = W[(size_t)(k0 + 2 * hi) * D + n0 + row];
        b.y = W[(size_t)(k0 + 2 * hi + 1) * D + n0 + row];
        acc = __builtin_amdgcn_wmma_f32_16x16x4_f32(false, a, false, b,
                                                    (short)0, acc, false, false);
    }
    const float bcol = bias[n0 + row];
    #pragma unroll
    for (int i = 0; i < 8; ++i) {
        // D layout: VGPR i, lanes 0-15 -> M=i, N=lane; lanes 16-31 -> M=8+i, N=lane-16
        H[(size_t)(m0 + 8 * hi + i) * D + n0 + row] = acc[i] + bcol;
    }
}

// ---------------- BN stats: partial column sums / sumsq ----------------
// block 256: thread t owns column t&127, row-parity t>>7; atomicAdd into sums[256]
__global__ void bn_partial_k(const float* __restrict__ H, float* __restrict__ sums,
                             int N, int rowsPerBlock) {
    __shared__ float ls[256], lq[256];
    const int col   = threadIdx.x & 127;
    const int rpair = threadIdx.x >> 7;
    const int r0    = blockIdx.x * rowsPerBlock;
    const int r1    = min(r0 + rowsPerBlock, N);
    float s = 0.0f, q = 0.0f;
    for (int r = r0 + rpair; r < r1; r += 2) {
        float v = H[(size_t)r * D + col];
        s += v; q += v * v;
    }
    ls[threadIdx.x] = s; lq[threadIdx.x] = q;
    __syncthreads();
    if (threadIdx.x < 128) {
        atomic_add_f32(&sums[threadIdx.x],       ls[threadIdx.x] + ls[threadIdx.x + 128]);
        atomic_add_f32(&sums[threadIdx.x + 128], lq[threadIdx.x] + lq[threadIdx.x + 128]);
    }
}

// ---------------- BN finalize: scale/shift ----------------
__global__ void bn_finalize_k(const float* __restrict__ sums, const float* __restrict__ gamma,
                              const float* __restrict__ beta, float* __restrict__ scsh, int N) {
    int t = threadIdx.x;
    if (t < 128) {
        float invN = 1.0f / (float)N;
        float mu   = sums[t] * invN;
        float var  = sums[t + 128] * invN - mu * mu;
        float rstd = rsqrtf(var + BN_EPS);
        float sc   = rstd * gamma[t];
        scsh[t]        = sc;                  // scale
        scsh[t + 128]  = beta[t] - mu * sc;   // shift
    }
}

// ---------------- GEMM2: H2 = relu(H1*scale+shift) @ W2 + b2 ----------------
// also writes hidden = temp0 * H2 (epilogue fusion)
__global__ void gemm2_k(const float* __restrict__ H1, const float* __restrict__ scsh,
                        const float* __restrict__ W, const float* __restrict__ bias,
                        float* __restrict__ H2, float* __restrict__ hidden, float temp0) {
    const int wave = threadIdx.x >> 5;
    const int lane = threadIdx.x & 31;
    const int row  = lane & 15;
    const int hi   = lane >> 4;
    const int m0   = blockIdx.x * 16;
    const int n0   = wave * 16;

    v8f acc = {};
    #pragma unroll
    for (int k0 = 0; k0 < D; k0 += 4) {
        const int kk = k0 + 2 * hi;
        v2f araw = *(const v2f*)&H1[(size_t)(m0 + row) * D + kk];
        v2f a;
        a.x = fmaxf(araw.x * scsh[kk]     + scsh[kk + 128],     0.0f);
        a.y = fmaxf(araw.y * scsh[kk + 1] + scsh[kk + 1 + 128], 0.0f);
        v2f b;
        b.x = W[(size_t)kk * D + n0 + row];
        b.y = W[(size_t)(kk + 1) * D + n0 + row];
        acc = __builtin_amdgcn_wmma_f32_16x16x4_f32(false, a, false, b,
                                                    (short)0, acc, false, false);
    }
    const float bcol = bias[n0 + row];
    #pragma unroll
    for (int i = 0; i < 8; ++i) {
        size_t idx = (size_t)(m0 + 8 * hi + i) * D + n0 + row;
        float v = acc[i] + bcol;
        H2[idx]     = v;
        hidden[idx] = temp0 * v;
    }
}

// ---------------- degree accumulation over E edges ----------------
__global__ void degree_k(const int* __restrict__ dst, float* __restrict__ deg, int E) {
    int e = blockIdx.x * blockDim.x + threadIdx.x;
    if (e < E) atomic_add_f32(&deg[dst[e]], 1.0f);
}

// dinv[i] = rsqrt(deg[i] + 1)  (self-loop included; in-place on deg)
__global__ void dinv_k(float* __restrict__ deg, int N) {
    int i = blockIdx.x * blockDim.x + threadIdx.x;
    if (i < N) deg[i] = rsqrtf(deg[i] + 1.0f);
}

// ---------------- edge scatter: next[dst] += dinv[src]*dinv[dst] * cur[src] ----------------
// one wave per edge; lane owns features [4*lane, 4*lane+3] (float4 gather, 4 atomics)
__global__ void scatter_k(const int* __restrict__ src, const int* __restrict__ dst,
                          const float* __restrict__ dinv, const float* __restrict__ cur,
                          float* __restrict__ next, int E) {
    int e = (blockIdx.x * blockDim.x + threadIdx.x) >> 5;
    if (e >= E) return;
    int lane = threadIdx.x & 31;
    int s = src[e];
    int d = dst[e];
    float w = dinv[s] * dinv[d];
    const float* cs = &cur[(size_t)s * D + 4 * lane];
    float* nd       = &next[(size_t)d * D + 4 * lane];
    float4 v = *(const float4*)cs;
    atomic_add_f32(nd + 0, w * v.x);
    atomic_add_f32(nd + 1, w * v.y);
    atomic_add_f32(nd + 2, w * v.z);
    atomic_add_f32(nd + 3, w * v.w);
}

// ---------------- fuse: self-loop add + finalize next + hidden AXPY ----------------
__global__ void fuse_k(const float* __restrict__ cur, float* __restrict__ next,
                       float* __restrict__ hidden, const float* __restrict__ dinv,
                       float coef, int ND) {
    int idx = blockIdx.x * blockDim.x + threadIdx.x;
    if (idx >= ND) return;
    int i = idx >> 7;                 // row
    float w  = dinv[i];
    float nv = next[idx] + w * w * cur[idx];
    next[idx] = nv;
    hidden[idx] += coef * nv;
}

// ---------------- final: out = x + relu(hidden) ----------------
__global__ void final_k(const float* __restrict__ x, const float* __restrict__ hidden,
                        float* __restrict__ out, int ND) {
    int idx = blockIdx.x * blockDim.x + threadIdx.x;
    if (idx < ND) out[idx] = x[idx] + fmaxf(hidden[idx], 0.0f);
}

extern "C" void kernel_launch(void* const* d_in, const int* in_sizes, int n_in,
                              void* d_out, int out_size, void* d_ws, size_t ws_size,
                              hipStream_t stream) {
    const float* x     = (const float*)d_in[0];
    const int*   edges = (const int*)d_in[1];
    const float* W1    = (const float*)d_in[2];
    const float* b1    = (const float*)d_in[3];
    const float* gamma = (const float*)d_in[4];
    const float* beta  = (const float*)d_in[5];
    const float* W2    = (const float*)d_in[6];
    const float* b2    = (const float*)d_in[7];
    float* out = (float*)d_out;

    const int N  = in_sizes[0] / D;      // 50000
    const int E  = in_sizes[1] / 2;      // 600000
    const int ND = N * D;
    const int* src = edges;
    const int* dst = edges + E;

    // GPR coefficients: temp[k] = 0.1 * 0.9^k (k<10), temp[10] = 0.9^10
    float temp[11];
    {
        float p = 1.0f;
        for (int k = 0; k < 10; ++k) { temp[k] = 0.1f * p; p *= 0.9f; }
        temp[10] = p;
    }

    // workspace layout (floats)
    float* ws     = (float*)d_ws;
    float* bufA   = ws;                  // h1, then ping-pong
    float* bufB   = ws + (size_t)ND;     // h2 = cur0, then ping-pong
    float* hidden = ws + 2 * (size_t)ND;
    float* deg    = ws + 3 * (size_t)ND; // N floats -> becomes dinv in place
    float* sums   = deg + N;             // 256 floats (sum | sumsq)
    float* scsh   = sums + 256;          // 256 floats (scale | shift)

    const int rowTiles = N / 16;         // 3125 (exact)

    // zero deg[N] + sums[256] (contiguous)
    zero_k<<<(N + 256 + 255) / 256, 256, 0, stream>>>(deg, N + 256);

    // MLP stage 1
    gemm1_k<<<rowTiles, 256, 0, stream>>>(x, W1, b1, bufA);

    // BatchNorm stats
    const int statBlocks = 256;
    const int rowsPerBlock = (N + statBlocks - 1) / statBlocks;
    bn_partial_k<<<statBlocks, 256, 0, stream>>>(bufA, sums, N, rowsPerBlock);
    bn_finalize_k<<<1, 128, 0, stream>>>(sums, gamma, beta, scsh, N);

    // MLP stage 2 (BN+ReLU fused on A loads), hidden = temp[0]*h2
    gemm2_k<<<rowTiles, 256, 0, stream>>>(bufA, scsh, W2, b2, bufB, hidden, temp[0]);

    // GCN normalization
    degree_k<<<(E + 255) / 256, 256, 0, stream>>>(dst, deg, E);
    dinv_k<<<(N + 255) / 256, 256, 0, stream>>>(deg, N);

    // K=10 propagation hops, ping-pong bufB (cur) <-> bufA (next)
    float* cur  = bufB;
    float* next = bufA;
    const int scatterBlocks = (E * 32 + 255) / 256;
    for (int k = 0; k < 10; ++k) {
        zero_k<<<(ND + 255) / 256, 256, 0, stream>>>(next, ND);
        scatter_k<<<scatterBlocks, 256, 0, stream>>>(src, dst, deg, cur, next, E);
        fuse_k<<<(ND + 255) / 256, 256, 0, stream>>>(cur, next, hidden, deg, temp[k + 1], ND);
        float* t = cur; cur = next; next = t;
    }

    // out = x + relu(hidden)
    final_k<<<(ND + 255) / 256, 256, 0, stream>>>(x, hidden, out, ND);
}